// Net_32195074851185
// MI455X (gfx1250) — compile-verified
//
#include <hip/hip_runtime.h>
#include <hip/hip_bf16.h>

// SNN forward: T=100, B=256, sizes 784 -> 64 -> 64 -> 10, fp32.
// Kernel 1: I0 = inp @ W0^T + b0 via V_WMMA_F32_16X16X4_F32 (91% of FLOPs, fully parallel).
// Kernel 2: sequential LIF recurrence; per-wave 16 batch rows, W1/W2 in LDS, WMMA matvecs.

typedef __attribute__((ext_vector_type(2))) float v2f;
typedef __attribute__((ext_vector_type(8))) float v8f;

__device__ __forceinline__ v8f wmma_f32x4(v2f a, v2f b, v8f c) {
  // 8 args: (neg_a, A, neg_b, B, c_mod, C, reuse_a, reuse_b)
  return __builtin_amdgcn_wmma_f32_16x16x4_f32(false, a, false, b, (short)0, c,
                                               false, false);
}

#define TT 100
#define BB 256
#define N0 784
#define H 64
#define NOUT 10
#define TB (TT * BB)                 // 25600 rows
#define LS_PER_T (BB * (H + H + NOUT)) // 35328
#define OUTSPK_TOTAL (TT * BB * NOUT)  // 256000

// ---------------------------------------------------------------------------
// Kernel 1: I0[r, n] = sum_k inp[r, k] * W0[n, k] + b0[n];  r in [0,25600)
// block = 128 threads = 4 waves; wave w -> output col tile w (cols w*16..w*16+15)
// blockIdx.x -> row tile (16 rows).
// A fragment (16x4 f32): lane<16 holds A[lid, k..k+1], lane>=16 holds A[lid, k+2..k+3]
// B fragment (4x16 f32): lane<16 holds B[k..k+1, lid] = W0[col, k..k+1] (contiguous)
// D tile: lane l: N = l%16, M = j + 8*(l/16) for vgpr j.
// ---------------------------------------------------------------------------
__global__ void __launch_bounds__(128) snn_gemm0(const float* __restrict__ inp,
                                                 const float* __restrict__ W0,
                                                 const float* __restrict__ b0,
                                                 float* __restrict__ I0) {
  const int lane = threadIdx.x & 31;
  const int wave = threadIdx.x >> 5;   // col tile 0..3
  const int half = lane >> 4;          // K-split (A/B) or M-split (D)
  const int lid  = lane & 15;
  const int r0   = blockIdx.x * 16;

  const float* aptr = inp + (size_t)(r0 + lid) * N0 + half * 2;
  const int bcol = wave * 16 + lid;
  const float* bptr = W0 + (size_t)bcol * N0 + half * 2;

  v8f acc = {};
#pragma unroll 4
  for (int k = 0; k < N0; k += 4) {
    v2f a = *(const v2f*)(aptr + k);
    v2f b = *(const v2f*)(bptr + k);
    acc = wmma_f32x4(a, b, acc);
  }

  const float bias = b0[bcol];
  float* out = I0 + (size_t)(r0 + 8 * half) * H + bcol;
#pragma unroll
  for (int j = 0; j < 8; ++j) out[(size_t)j * H] = acc[j] + bias;
}

// ---------------------------------------------------------------------------
// Kernel 2: LIF recurrence over T. grid = 16 blocks x 32 threads (1 wave).
// Each wave owns batch rows [r0, r0+16). v-state lives in registers in the
// WMMA C/D layout. Spikes go through LDS scratch (row-major 16x64) so they can
// be re-loaded as WMMA A fragments.
// ---------------------------------------------------------------------------
__global__ void __launch_bounds__(32) snn_recur(const float* __restrict__ I0,
                                                const float* __restrict__ W1,
                                                const float* __restrict__ b1,
                                                const float* __restrict__ W2,
                                                const float* __restrict__ b2,
                                                float* __restrict__ out) {
  __shared__ float sW1[H * H];     // W1[n, k], row-major (B operand: contiguous K pairs)
  __shared__ float sW2[16 * H];    // W2 zero-padded to 16 output cols
  __shared__ float sZ[16 * H];     // spike scratch, row-major [m, k]

  const int lane = threadIdx.x;    // 0..31
  const int half = lane >> 4;
  const int lid  = lane & 15;
  const int r0   = blockIdx.x * 16;

  for (int i = lane; i < H * H; i += 32) sW1[i] = W1[i];
  for (int i = lane; i < 16 * H; i += 32) sW2[i] = (i < NOUT * H) ? W2[i] : 0.0f;
  __syncthreads();

  float bias1[4];
#pragma unroll
  for (int n = 0; n < 4; ++n) bias1[n] = b1[n * 16 + lid];
  const float bias2 = (lid < NOUT) ? b2[lid] : 0.0f;

  v8f v0[4] = {};   // 16 rows x 64 cols, D layout, 4 col tiles
  v8f v1[4] = {};
  v8f v2t  = {};    // 16 rows x 16 cols (cols >= 10 stay zero)

  float* outspk = out;                       // [T, B, 10]
  float* lsall  = out + OUTSPK_TOTAL;        // [T, 35328]

  for (int t = 0; t < TT; ++t) {
    const float* i0p =
        I0 + ((size_t)t * BB + r0 + 8 * half) * H + lid;   // + j*H + n*16
    float* ls = lsall + (size_t)t * LS_PER_T;

    // ---- layer 0: LIF on precomputed currents ----
#pragma unroll
    for (int n = 0; n < 4; ++n) {
#pragma unroll
      for (int j = 0; j < 8; ++j) {
        float ic = i0p[(size_t)j * H + n * 16];
        float v  = v0[n][j];
        v = v + 0.05f * (ic - v);                  // leaky integrate
        float z = ((v - 1.0f) > 0.0f) ? 1.0f : 0.0f;
        v0[n][j] = v - z;                          // reset_subtract
        const int m = j + 8 * half;
        const int col = n * 16 + lid;
        sZ[m * H + col] = z;
        ls[(size_t)(r0 + m) * H + col] = z;        // layer_spikes z0
      }
    }
    __syncthreads();

    // ---- layer 1: i1 = z0 @ W1^T (16x64 @ 64x64) ----
    v8f acc[4] = {};
    for (int k = 0; k < H; k += 4) {
      v2f a = *(const v2f*)(&sZ[lid * H + k + half * 2]);
#pragma unroll
      for (int n = 0; n < 4; ++n) {
        v2f b = *(const v2f*)(&sW1[(n * 16 + lid) * H + k + half * 2]);
        acc[n] = wmma_f32x4(a, b, acc[n]);
      }
    }
    __syncthreads();   // done reading z0 scratch

    // ---- layer 1 LIF ----
#pragma unroll
    for (int n = 0; n < 4; ++n) {
#pragma unroll
      for (int j = 0; j < 8; ++j) {
        float ic = acc[n][j] + bias1[n];
        float v  = v1[n][j];
        v = v + 0.05f * (ic - v);
        float z = ((v - 1.0f) > 0.0f) ? 1.0f : 0.0f;
        v1[n][j] = v - z;
        const int m = j + 8 * half;
        const int col = n * 16 + lid;
        sZ[m * H + col] = z;
        ls[(size_t)(BB * H) + (size_t)(r0 + m) * H + col] = z;  // z1
      }
    }
    __syncthreads();

    // ---- layer 2: i2 = z1 @ W2^T (16x64 @ 64x16, cols>=10 are zero) ----
    v8f acc2 = {};
    for (int k = 0; k < H; k += 4) {
      v2f a = *(const v2f*)(&sZ[lid * H + k + half * 2]);
      v2f b = *(const v2f*)(&sW2[lid * H + k + half * 2]);
      acc2 = wmma_f32x4(a, b, acc2);
    }
    __syncthreads();

    // ---- layer 2 LIF + outputs ----
#pragma unroll
    for (int j = 0; j < 8; ++j) {
      float ic = acc2[j] + bias2;
      float v  = v2t[j];
      v = v + 0.05f * (ic - v);
      float z = ((v - 1.0f) > 0.0f) ? 1.0f : 0.0f;
      v2t[j] = v - z;
      if (lid < NOUT) {
        const int m = j + 8 * half;
        outspk[((size_t)t * BB + r0 + m) * NOUT + lid] = z;               // out_spikes
        ls[(size_t)(2 * BB * H) + (size_t)(r0 + m) * NOUT + lid] = z;     // z2
      }
    }
  }
}

extern "C" void kernel_launch(void* const* d_in, const int* in_sizes, int n_in,
                              void* d_out, int out_size, void* d_ws, size_t ws_size,
                              hipStream_t stream) {
  const float* inp = (const float*)d_in[0];
  const float* W0  = (const float*)d_in[1];
  const float* b0  = (const float*)d_in[2];
  const float* W1  = (const float*)d_in[3];
  const float* b1  = (const float*)d_in[4];
  const float* W2  = (const float*)d_in[5];
  const float* b2  = (const float*)d_in[6];
  float* out = (float*)d_out;
  float* I0  = (float*)d_ws;   // 25600 x 64 f32 = 6.55 MB scratch

  snn_gemm0<<<TB / 16, 128, 0, stream>>>(inp, W0, b0, I0);
  snn_recur<<<BB / 16, 32, 0, stream>>>(I0, W1, b1, W2, b2, out);
}